// RNN_24077586662061
// MI455X (gfx1250) — compile-verified
//
#include <hip/hip_runtime.h>
#include <hip/hip_bf16.h>
#include <math.h>

// ---------------------------------------------------------------------------
// Two masked tanh-RNN scans (d=512) over B=2048 rows + 2-col readout.
// h' = tanh(W_h h + ctx*w_ctx + b), per-row freeze after steps[row] iters.
//
// Rows are independent -> no grid sync. Each workgroup (16 waves) owns TWO
// 16-row tiles resident in LDS and runs the full recurrence locally with
// V_WMMA_F32_16X16X4_F32 (fp32 matrix path = reference precision).
// Each wave owns 32 output columns and reuses every W fragment (loaded from
// L2-resident packed W) across both row tiles in registers -> halves L2
// traffic vs 1-tile/WG and gives 2048 waves of parallelism.
// Freezing = branchless skip-the-store: frozen elements write to a dump slot
// (v_cndmask on the LDS address instead of saveexec branches).
// ---------------------------------------------------------------------------

#define D_MODEL 512
#define BATCH   2048
#define MAXP    150
#define MAXN    200
#define TP      143      // MAX_P - MIN_P + 1
#define TN      193      // MAX_N - MIN_N + 1
#define HSTRIDE 516      // 512 + 4 pad: bank = (4*m + k) mod 64 -> conflict-free
#define TILEF   (16 * HSTRIDE)   // floats per 16-row tile
#define DUMPOFF (2 * TILEF)      // dump slots for frozen-row stores

typedef __attribute__((ext_vector_type(2))) float v2f;
typedef __attribute__((ext_vector_type(8))) float v8f;

__device__ __forceinline__ float fast_tanh(float x) {
#if __has_builtin(__builtin_amdgcn_tanhf)
  return __builtin_amdgcn_tanhf(x);   // gfx1250 v_tanh_f32
#else
  return tanhf(x);
#endif
}

// Repack W (d x (d+1) row-major) into B-fragment order for 16x16x4 f32 WMMA:
// Wpack[((kt*512) + n)*4 + kk] = W[n*513 + kt*4 + kk]
// -> each lane's b64 fragment load is 8-byte aligned (raw stride 513 is odd).
__global__ void pack_w_kernel(const float* __restrict__ W,
                              float* __restrict__ Wpack) {
  int o  = blockIdx.x * blockDim.x + threadIdx.x;   // 512*512 elements
  int kk = o & 3;
  int n  = (o >> 2) & 511;
  int kt = o >> 11;
  Wpack[o] = W[n * 513 + kt * 4 + kk];
}

// Workgroup = 512 threads = 16 waves. WG owns 32 rows (2 tiles of 16 in LDS);
// wave w owns columns [w*32, w*32+32). blockIdx.x = row block (0..63),
// blockIdx.y = scan (0 = protons, 1 = neutrons).
__global__ __launch_bounds__(512)
void rnn_scan_kernel(const int*   __restrict__ x,        // [B,2] (n_p, n_n)
                     const float* __restrict__ emb,      // [4,512]
                     const float* __restrict__ Wp_raw,
                     const float* __restrict__ bp,
                     const float* __restrict__ Wn_raw,
                     const float* __restrict__ bn,
                     const float* __restrict__ WpPack,
                     const float* __restrict__ WnPack,
                     float*       __restrict__ hp_out,   // [B,512]
                     float*       __restrict__ hn_out)   // [B,512]
{
  __shared__ float Hs[2 * TILEF + 32];

  const int scan    = blockIdx.y;
  const int rowbase = blockIdx.x * 32;
  const int tid     = threadIdx.x;
  const int lane    = tid & 31;
  const int wv      = tid >> 5;        // 0..15: column block
  const int hl      = lane >> 4;       // lane half
  const int ln      = lane & 15;
  const int cb      = wv * 32;

  const float *Wraw, *bvec, *Wpk, *h0;
  float* hout;
  int T, ctx_col, steps_col;
  float ctx_scale;
  if (scan == 0) { // protons: ctx = n_n/MAX_N, steps = n_p - 8 + 1
    Wraw = Wp_raw; bvec = bp; Wpk = WpPack; hout = hp_out;
    h0 = emb;           T = TP; ctx_col = 1; steps_col = 0; ctx_scale = 1.0f / MAXN;
  } else {         // neutrons: ctx = n_p/MAX_P, steps = n_n - 8 + 1
    Wraw = Wn_raw; bvec = bn; Wpk = WnPack; hout = hn_out;
    h0 = emb + D_MODEL; T = TN; ctx_col = 0; steps_col = 1; ctx_scale = 1.0f / MAXP;
  }

  // ---- init both H tiles: every row = embeddings[scan] ----
  for (int idx = tid; idx < 2 * 16 * D_MODEL; idx += 512) {
    int tile = idx >> 13, r = (idx >> 9) & 15, c = idx & 511;
    Hs[tile * TILEF + r * HSTRIDE + c] = h0[c];
  }

  // ---- per-lane row metadata: C/D element i <-> row m = i + 8*hl ----
  int   steps_lane[2][8];
  float ctxv[2][8];
  int   tmax = 1;
  #pragma unroll
  for (int rt = 0; rt < 2; ++rt) {
    #pragma unroll
    for (int i = 0; i < 8; ++i) {
      int gr = rowbase + rt * 16 + i + 8 * hl;
      ctxv[rt][i] = (float)x[gr * 2 + ctx_col] * ctx_scale;
      int st = x[gr * 2 + steps_col] - 8 + 1;   // MIN = 8
      if (st < 1) st = 1;
      if (st > T) st = T;
      steps_lane[rt][i] = st;
      tmax = st > tmax ? st : tmax;
    }
  }
  // halves hold different rows; all waves see the same 32 rows, so every
  // wave derives the identical loop bound (barrier counts must match).
  tmax = max(tmax, __shfl_xor(tmax, 16, 32));

  // ---- step-invariant C fragments: C[m,j] = ctx[m]*w_ctx[col] + b[col] ----
  v8f cfrag[2][2];
  #pragma unroll
  for (int j = 0; j < 2; ++j) {
    int col  = cb + j * 16 + ln;
    float wc = Wraw[col * 513 + 512];
    float bb = bvec[col];
    #pragma unroll
    for (int rt = 0; rt < 2; ++rt)
      #pragma unroll
      for (int i = 0; i < 8; ++i) cfrag[rt][j][i] = ctxv[rt][i] * wc + bb;
  }
  __syncthreads();

  // ---- recurrence ----
  for (int t = 0; t < tmax; ++t) {
    v8f acc[2][2];
    #pragma unroll
    for (int rt = 0; rt < 2; ++rt)
      #pragma unroll
      for (int j = 0; j < 2; ++j) acc[rt][j] = cfrag[rt][j];

    #pragma unroll 4
    for (int kt = 0; kt < 128; ++kt) {
      // A fragments: 16x4 of H. lanes0-15: K=k0,k0+1 ; lanes16-31: K=k0+2,k0+3
      v2f a0 = *(const v2f*)&Hs[0 * TILEF + ln * HSTRIDE + kt * 4 + 2 * hl];
      v2f a1 = *(const v2f*)&Hs[1 * TILEF + ln * HSTRIDE + kt * 4 + 2 * hl];
      #pragma unroll
      for (int j = 0; j < 2; ++j) {
        int n = cb + j * 16 + ln;
        v2f b = *(const v2f*)&Wpk[(kt * 512 + n) * 4 + 2 * hl];
        // one L2 fragment feeds both row tiles (register reuse)
        acc[0][j] = __builtin_amdgcn_wmma_f32_16x16x4_f32(
            false, a0, false, b, (short)0, acc[0][j], false, false);
        acc[1][j] = __builtin_amdgcn_wmma_f32_16x16x4_f32(
            false, a1, false, b, (short)0, acc[1][j], false, false);
      }
    }

    __syncthreads();   // all reads of Hs complete before in-place update

    // Branchless freeze: frozen elements store into the dump slot instead of
    // their H location (address cndmask; no exec-mask branching).
    #pragma unroll
    for (int rt = 0; rt < 2; ++rt) {
      #pragma unroll
      for (int j = 0; j < 2; ++j) {
        int col = cb + j * 16 + ln;
        #pragma unroll
        for (int i = 0; i < 8; ++i) {
          int m    = i + 8 * hl;
          int addr = (t < steps_lane[rt][i])
                       ? (rt * TILEF + m * HSTRIDE + col)
                       : (DUMPOFF + lane);
          Hs[addr] = fast_tanh(acc[rt][j][i]);
        }
      }
    }
    __syncthreads();
  }

  // ---- write final 32 rows x 32 cols (this wave's column block) ----
  for (int idx = lane; idx < 32 * 32; idx += 32) {
    int r = idx >> 5, c = idx & 31;   // lanes contiguous in c -> coalesced
    hout[(rowbase + r) * D_MODEL + cb + c] =
        Hs[(r >> 4) * TILEF + (r & 15) * HSTRIDE + cb + c];
  }
}

// out[row, o] = sum_c hp[row,c]*Wr[o,c] + hn[row,c]*Wr[o,512+c] + br[o]
__global__ __launch_bounds__(256)
void readout_kernel(const float* __restrict__ hp, const float* __restrict__ hn,
                    const float* __restrict__ Wr, const float* __restrict__ br,
                    float* __restrict__ out)
{
  int wv   = threadIdx.x >> 5;
  int lane = threadIdx.x & 31;
  int row  = blockIdx.x * 8 + wv;
  if (row >= BATCH) return;
  float s0 = 0.f, s1 = 0.f;
  for (int c = lane; c < 512; c += 32) {
    float a = hp[row * 512 + c];
    float b = hn[row * 512 + c];
    s0 += a * Wr[c]        + b * Wr[512 + c];
    s1 += a * Wr[1024 + c] + b * Wr[1536 + c];
  }
  #pragma unroll
  for (int off = 16; off; off >>= 1) {
    s0 += __shfl_xor(s0, off, 32);
    s1 += __shfl_xor(s1, off, 32);
  }
  if (lane == 0) {
    out[row * 2 + 0] = s0 + br[0];
    out[row * 2 + 1] = s1 + br[1];
  }
}

extern "C" void kernel_launch(void* const* d_in, const int* in_sizes, int n_in,
                              void* d_out, int out_size, void* d_ws, size_t ws_size,
                              hipStream_t stream) {
  const int*   x   = (const int*)  d_in[0];   // [B,2]
  const float* emb = (const float*)d_in[1];   // [4,512]
  const float* Wp  = (const float*)d_in[2];   // [512,513]
  const float* bp  = (const float*)d_in[3];
  const float* Wn  = (const float*)d_in[4];
  const float* bn  = (const float*)d_in[5];
  const float* Wr  = (const float*)d_in[6];   // [2,1024]
  const float* br  = (const float*)d_in[7];
  float* out = (float*)d_out;

  // workspace layout (floats): hp[1M] | hn[1M] | WpPack[256K] | WnPack[256K] (~10 MB)
  float* ws     = (float*)d_ws;
  float* hp     = ws;
  float* hn     = ws + (size_t)BATCH * D_MODEL;
  float* WpPack = ws + (size_t)2 * BATCH * D_MODEL;
  float* WnPack = WpPack + (size_t)D_MODEL * D_MODEL;

  pack_w_kernel<<<(D_MODEL * D_MODEL) / 256, 256, 0, stream>>>(Wp, WpPack);
  pack_w_kernel<<<(D_MODEL * D_MODEL) / 256, 256, 0, stream>>>(Wn, WnPack);

  rnn_scan_kernel<<<dim3(BATCH / 32, 2), 512, 0, stream>>>(
      x, emb, Wp, bp, Wn, bn, WpPack, WnPack, hp, hn);

  readout_kernel<<<BATCH / 8, 256, 0, stream>>>(hp, hn, Wr, br, out);
}